// myGCN_34900904247516
// MI455X (gfx1250) — compile-verified
//
#include <hip/hip_runtime.h>
#include <hip/hip_bf16.h>

// ---------------------------------------------------------------------------
// GCN forward on MI455X (gfx1250, wave32).
//
// Roofline: ~13 GFLOP total, ~20 MB working set (fits in 192 MB L2) ->
// latency/launch bound, not HBM bound. We use the native fp32 matrix pipe
// (V_WMMA_F32_16X16X4_F32) to stay bit-faithful to the fp32 reference;
// f16/fp8 peak cannot be utilized by a problem this small.
//
// Each wave computes a 16x64 output strip (4 N-tiles): the A fragment is
// loaded once per K-step and fed to 4 back-to-back WMMAs -> 4x less A
// traffic, 4x less address VALU per WMMA.
//
// WMMA fragment layouts (CDNA5 ISA 7.12.2, wave32):
//   A (16x4 f32, 2 VGPRs/lane):  M = lane%16 ; lanes 0-15 hold K={k0,k0+1},
//                                lanes 16-31 hold K={k0+2,k0+3}  -> float2 load
//   B (4x16 f32, 2 VGPRs):       N = lane%16 ; VGPR r holds row k0+r (lanes
//                                0-15) and row k0+r+2 (lanes 16-31)
//   C/D (16x16 f32, 8 VGPRs):    N = lane%16 ; VGPR r holds row 16*mt+r
//                                (lanes 0-15) / +8 (lanes 16-31)
// ---------------------------------------------------------------------------

#define BB 8
#define LL 256
#define DD 512
#define EE 64

typedef __attribute__((ext_vector_type(2))) float v2f;
typedef __attribute__((ext_vector_type(8))) float v8f;

__device__ __forceinline__ v8f wmma4f32(v2f a, v2f b, v8f c) {
    // 8 args: (neg_a, A, neg_b, B, c_mod, C, reuse_a, reuse_b)
    return __builtin_amdgcn_wmma_f32_16x16x4_f32(
        false, a, false, b, (short)0, c, false, false);
}

// --------------------------------------------------------------------------
// esum[b,i,:] = sum_j adj[b,i,j] * emb[edge[b,i,j],:]   (iteration-invariant)
// invn[b,i]  = 1 / (#(adj[b,i,:]>0) + eps)
// One 64-thread block per (b,i) row; each thread owns one of the E=64 lanes.
// --------------------------------------------------------------------------
__global__ void esum_kernel(const float* __restrict__ adj,
                            const int*   __restrict__ edge,
                            const float* __restrict__ emb,
                            float* __restrict__ esum,
                            float* __restrict__ invn) {
    const int row = blockIdx.x;          // b*L + i
    const int t   = threadIdx.x;         // 0..63
    const float* arow = adj  + (size_t)row * LL;
    const int*   erow = edge + (size_t)row * LL;
    float acc = 0.f, cnt = 0.f;
    for (int j = 0; j < LL; ++j) {
        const float a = arow[j];
        if (a > 0.f) {                   // block-uniform branch
            acc += a * emb[(size_t)erow[j] * EE + t];
            cnt += 1.f;
        }
    }
    esum[(size_t)row * EE + t] = acc;
    if (t == 0) invn[row] = 1.f / (cnt + 1e-10f);
}

// --------------------------------------------------------------------------
// hagg[b] = adj[b](LxL) @ hid[b](LxD).  16x64 strip per wave, K-step 4.
// grid = (16, B), block = 256 (8 waves): 128 strips/batch.
// --------------------------------------------------------------------------
__global__ void hagg_kernel(const float* __restrict__ adj,
                            const float* __restrict__ hid,
                            float* __restrict__ hagg) {
    const int b     = blockIdx.y;
    const int wave  = threadIdx.x >> 5;
    const int lane  = threadIdx.x & 31;
    const int strip = blockIdx.x * 8 + wave;     // 0..127
    const int mt = strip >> 3;                   // D/64 = 8 strips
    const int nb = (strip & 7) * 64;
    const int ml = lane & 15;
    const int m  = mt * 16 + ml;
    const int koff = (lane >= 16) ? 2 : 0;

    const float* A  = adj + (size_t)b * LL * LL + (size_t)m * LL;
    const float* Bm = hid + (size_t)b * LL * DD;

    v8f c0 = {}, c1 = {}, c2 = {}, c3 = {};
    for (int k0 = 0; k0 < LL; k0 += 4) {
        v2f a = *(const v2f*)(A + k0 + koff);           // shared by 4 WMMAs
        const float* br = Bm + (size_t)(k0 + koff) * DD + nb + ml;
        v2f b0, b1, b2, b3;
        b0.x = br[0];  b0.y = br[DD + 0];
        b1.x = br[16]; b1.y = br[DD + 16];
        b2.x = br[32]; b2.y = br[DD + 32];
        b3.x = br[48]; b3.y = br[DD + 48];
        c0 = wmma4f32(a, b0, c0);
        c1 = wmma4f32(a, b1, c1);
        c2 = wmma4f32(a, b2, c2);
        c3 = wmma4f32(a, b3, c3);
    }

    float* C = hagg + (size_t)b * LL * DD;
    const int mrow = mt * 16 + ((lane >= 16) ? 8 : 0);
#pragma unroll
    for (int r = 0; r < 8; ++r) {
        float* cr = C + (size_t)(mrow + r) * DD + nb + ml;
        cr[0] = c0[r]; cr[16] = c1[r]; cr[32] = c2[r]; cr[48] = c3[r];
    }
}

// --------------------------------------------------------------------------
// he = tanh( (cat[hagg, esum] * invn_row) @ W1^T ),  K = D+E = 576, N = E = 64.
// One wave computes a full 16x64 row block. grid = (2, B), block = 256.
// Row scale 1/innum folds into the A fragment (M = lane%16 -> per-lane scalar).
// --------------------------------------------------------------------------
__global__ void edgemlp_kernel(const float* __restrict__ hagg,  // (B,L,D)
                               const float* __restrict__ esum,  // (B,L,E)
                               const float* __restrict__ invn,  // (B,L)
                               const float* __restrict__ W1,    // (E, D+E)
                               float* __restrict__ he) {        // (B,L,E)
    const int b    = blockIdx.y;
    const int wave = threadIdx.x >> 5;
    const int lane = threadIdx.x & 31;
    const int mt   = blockIdx.x * 8 + wave;      // 0..15
    const int ml   = lane & 15;
    const int m    = mt * 16 + ml;
    const int koff = (lane >= 16) ? 2 : 0;
    const int KT   = DD + EE;

    const float inv = invn[b * LL + m];
    const float* ah = hagg + ((size_t)b * LL + m) * DD;
    const float* ae = esum + ((size_t)b * LL + m) * EE;
    const float* w0 = W1 + (size_t)(ml)      * KT;
    const float* w1 = W1 + (size_t)(ml + 16) * KT;
    const float* w2 = W1 + (size_t)(ml + 32) * KT;
    const float* w3 = W1 + (size_t)(ml + 48) * KT;

    v8f c0 = {}, c1 = {}, c2 = {}, c3 = {};
    for (int k0 = 0; k0 < KT; k0 += 4) {
        const int k = k0 + koff;
        v2f a = (k < DD) ? *(const v2f*)(ah + k)
                         : *(const v2f*)(ae + (k - DD));
        a *= inv;
        v2f b0 = *(const v2f*)(w0 + k);
        v2f b1 = *(const v2f*)(w1 + k);
        v2f b2 = *(const v2f*)(w2 + k);
        v2f b3 = *(const v2f*)(w3 + k);
        c0 = wmma4f32(a, b0, c0);
        c1 = wmma4f32(a, b1, c1);
        c2 = wmma4f32(a, b2, c2);
        c3 = wmma4f32(a, b3, c3);
    }

    float* C = he + (size_t)b * LL * EE;
    const int mrow = mt * 16 + ((lane >= 16) ? 8 : 0);
#pragma unroll
    for (int r = 0; r < 8; ++r) {
        float* cr = C + (size_t)(mrow + r) * EE + ml;
        cr[0]  = tanhf(c0[r]); cr[16] = tanhf(c1[r]);
        cr[32] = tanhf(c2[r]); cr[48] = tanhf(c3[r]);
    }
}

// --------------------------------------------------------------------------
// out = tanh( cat[A1, A2] @ W^T + bias ).  Serves W2/W4 (K1=64,K2=512) and
// W5 (K1=K2=512). 16x64 strip per wave; N = 512 -> 8 strips x 16 M-tiles.
// grid = (16, B), block = 256. K1 multiple of 4 -> float2 loads never
// straddle the concat boundary.
// --------------------------------------------------------------------------
__global__ void concatmlp_kernel(const float* __restrict__ A1,   // (B,L,K1)
                                 const float* __restrict__ A2,   // (B,L,K2)
                                 const int K1, const int K2,
                                 const float* __restrict__ W,    // (N, K1+K2)
                                 const float* __restrict__ bias, // (N)
                                 float* __restrict__ out,        // (B,L,N)
                                 const int N) {
    const int b     = blockIdx.y;
    const int wave  = threadIdx.x >> 5;
    const int lane  = threadIdx.x & 31;
    const int nstr  = N >> 6;                    // strips of 64
    const int strip = blockIdx.x * 8 + wave;
    const int mt = strip / nstr;
    const int nb = (strip % nstr) * 64;
    const int ml = lane & 15;
    const int m  = mt * 16 + ml;
    const int koff = (lane >= 16) ? 2 : 0;
    const int KT = K1 + K2;

    const float* a1 = A1 + ((size_t)b * LL + m) * K1;
    const float* a2 = A2 + ((size_t)b * LL + m) * K2;
    const float* w0 = W + (size_t)(nb + ml)      * KT;
    const float* w1 = W + (size_t)(nb + ml + 16) * KT;
    const float* w2 = W + (size_t)(nb + ml + 32) * KT;
    const float* w3 = W + (size_t)(nb + ml + 48) * KT;

    v8f c0 = {}, c1 = {}, c2 = {}, c3 = {};
    for (int k0 = 0; k0 < KT; k0 += 4) {
        const int k = k0 + koff;
        v2f a = (k < K1) ? *(const v2f*)(a1 + k)
                         : *(const v2f*)(a2 + (k - K1));
        v2f b0 = *(const v2f*)(w0 + k);
        v2f b1 = *(const v2f*)(w1 + k);
        v2f b2 = *(const v2f*)(w2 + k);
        v2f b3 = *(const v2f*)(w3 + k);
        c0 = wmma4f32(a, b0, c0);
        c1 = wmma4f32(a, b1, c1);
        c2 = wmma4f32(a, b2, c2);
        c3 = wmma4f32(a, b3, c3);
    }

    const float bv0 = bias[nb + ml];
    const float bv1 = bias[nb + ml + 16];
    const float bv2 = bias[nb + ml + 32];
    const float bv3 = bias[nb + ml + 48];
    float* C = out + (size_t)b * LL * N;
    const int mrow = mt * 16 + ((lane >= 16) ? 8 : 0);
#pragma unroll
    for (int r = 0; r < 8; ++r) {
        float* cr = C + (size_t)(mrow + r) * N + nb + ml;
        cr[0]  = tanhf(c0[r] + bv0); cr[16] = tanhf(c1[r] + bv1);
        cr[32] = tanhf(c2[r] + bv2); cr[48] = tanhf(c3[r] + bv3);
    }
}

// out2[b,:] = fused[b, mains[b], :]
__global__ void gather_kernel(const float* __restrict__ fused,
                              const int* __restrict__ mains,
                              float* __restrict__ out2) {
    const int b = blockIdx.x;
    const int d = threadIdx.x;               // 512 threads
    const int m = mains[b];
    out2[(size_t)b * DD + d] = fused[((size_t)b * LL + m) * DD + d];
}

extern "C" void kernel_launch(void* const* d_in, const int* in_sizes, int n_in,
                              void* d_out, int out_size, void* d_ws, size_t ws_size,
                              hipStream_t stream) {
    (void)in_sizes; (void)n_in; (void)out_size; (void)ws_size;

    const float* inputs = (const float*)d_in[0];
    const int*   mains  = (const int*)  d_in[1];
    const float* adj1   = (const float*)d_in[2];
    const float* adj2   = (const float*)d_in[3];
    const int*   edge1  = (const int*)  d_in[4];
    const int*   edge2  = (const int*)  d_in[5];
    const float* emb    = (const float*)d_in[6];
    const float* W1     = (const float*)d_in[7];
    const float* W2     = (const float*)d_in[8];
    const float* b2     = (const float*)d_in[9];
    const float* W3     = (const float*)d_in[10];
    const float* W4     = (const float*)d_in[11];
    const float* b4     = (const float*)d_in[12];
    const float* W5     = (const float*)d_in[13];
    const float* b5     = (const float*)d_in[14];
    float* out = (float*)d_out;

    // Workspace carve-up (~22.6 MB of fp32)
    float* ws = (float*)d_ws;
    size_t off = 0;
    const size_t HID = (size_t)BB * LL * DD;
    const size_t ESZ = (size_t)BB * LL * EE;
    float* hid1a = ws + off; off += HID;
    float* hid1b = ws + off; off += HID;
    float* hid2a = ws + off; off += HID;
    float* hid2b = ws + off; off += HID;
    float* hagg  = ws + off; off += HID;     // shared scratch (streams run serially)
    float* he    = ws + off; off += ESZ;     // shared scratch
    float* esum1 = ws + off; off += ESZ;
    float* esum2 = ws + off; off += ESZ;
    float* invn1 = ws + off; off += (size_t)BB * LL;
    float* invn2 = ws + off; off += (size_t)BB * LL;

    // hid1 = hid2 = inputs
    hipMemcpyAsync(hid1a, inputs, sizeof(float) * HID, hipMemcpyDeviceToDevice, stream);
    hipMemcpyAsync(hid2a, inputs, sizeof(float) * HID, hipMemcpyDeviceToDevice, stream);

    // Iteration-invariant edge sums + neighbor counts
    esum_kernel<<<BB * LL, EE, 0, stream>>>(adj1, edge1, emb, esum1, invn1);
    esum_kernel<<<BB * LL, EE, 0, stream>>>(adj2, edge2, emb, esum2, invn2);

    const dim3 blk(256);
    float* h1 = hid1a; float* h1n = hid1b;
    float* h2 = hid2a; float* h2n = hid2b;

    for (int it = 0; it < 3; ++it) {
        // stream 1
        hagg_kernel     <<<dim3(16, BB), blk, 0, stream>>>(adj1, h1, hagg);
        edgemlp_kernel  <<<dim3(2,  BB), blk, 0, stream>>>(hagg, esum1, invn1, W1, he);
        concatmlp_kernel<<<dim3(16, BB), blk, 0, stream>>>(he, h1, EE, DD, W2, b2, h1n, DD);
        { float* t = h1; h1 = h1n; h1n = t; }

        // stream 2
        hagg_kernel     <<<dim3(16, BB), blk, 0, stream>>>(adj2, h2, hagg);
        edgemlp_kernel  <<<dim3(2,  BB), blk, 0, stream>>>(hagg, esum2, invn2, W3, he);
        concatmlp_kernel<<<dim3(16, BB), blk, 0, stream>>>(he, h2, EE, DD, W4, b4, h2n, DD);
        { float* t = h2; h2 = h2n; h2n = t; }
    }

    // fused = tanh(cat[hid1, hid2] @ W5^T + b5) -> d_out[0 : B*L*D]
    concatmlp_kernel<<<dim3(16, BB), blk, 0, stream>>>(h1, h2, DD, DD, W5, b5, out, DD);

    // d_out[B*L*D : ] = fused[b, mains[b], :]
    gather_kernel<<<BB, DD, 0, stream>>>(out, mains, out + (size_t)BB * LL * DD);
}